// STNSampler_71708773974754
// MI455X (gfx1250) — compile-verified
//
#include <hip/hip_runtime.h>
#include <hip/hip_bf16.h>

// Problem constants (match reference: B=16, C=3, H=W=1024, N=512, PATCH=32)
#define PB     16
#define PC     3
#define PH     1024
#define PW     1024
#define PN     512
#define PATCH  32
#define HALF   16

// One block per patch. 256 threads; 3072 output floats per patch = 768 float4
// groups; each thread produces 3 float4 groups. Consecutive lanes cover
// consecutive 16B column groups -> coalesced 128-bit stores, and the 4 source
// loads per group are contiguous within a row -> coalesced gathers. OOB
// (zero-pad) handled with unsigned-compare predication per element.
__global__ __launch_bounds__(256)
void stn_patch_gather(const float* __restrict__ img,   // last image [C,H,W]
                      const int*   __restrict__ kps,   // last batch [N,2] (x,y)
                      float*       __restrict__ out)   // [N,C,32,32]
{
    const int n = blockIdx.x;           // patch index 0..511
    const int t = threadIdx.x;          // 0..255

    // (x, y) for this patch — same for all lanes; compiler scalarizes.
    const int x    = kps[n * 2 + 0];
    const int y    = kps[n * 2 + 1];
    const int left = x - HALF;
    const int top  = y - HALF;

    float* out_patch = out + (size_t)n * (PC * PATCH * PATCH);

#pragma unroll
    for (int k = 0; k < 3; ++k) {
        const int g    = t + (k << 8);          // float4 group 0..767
        const int c    = g >> 8;                // channel 0..2
        const int rem  = g & 255;
        const int r    = rem >> 3;              // patch row 0..31
        const int col0 = (rem & 7) << 2;        // patch col {0,4,...,28}

        const int  sy    = top + r;
        const bool rowin = (unsigned)sy < (unsigned)PH;

        const float* src_row = img + (size_t)c * (PH * PW) + (size_t)sy * PW;

        // gfx1250 prefetch: pull the source cachelines for this row segment
        // ahead of the four predicated scalar loads (emits global_prefetch_b8;
        // speculative — invalid addresses are silently dropped).
        if (rowin) {
            __builtin_prefetch(src_row + (left + col0), 0, 0);
        }

        float4 v;
        float* vv = reinterpret_cast<float*>(&v);
#pragma unroll
        for (int j = 0; j < 4; ++j) {
            const int  sx = left + col0 + j;
            const bool in = rowin && ((unsigned)sx < (unsigned)PW);
            vv[j] = in ? src_row[sx] : 0.0f;
        }

        // 16B-aligned, fully coalesced store: global_store_b128 per lane.
        *reinterpret_cast<float4*>(out_patch + c * (PATCH * PATCH)
                                   + r * PATCH + col0) = v;
    }
}

extern "C" void kernel_launch(void* const* d_in, const int* in_sizes, int n_in,
                              void* d_out, int out_size, void* d_ws, size_t ws_size,
                              hipStream_t stream) {
    const float* p   = (const float*)d_in[0];   // [16,3,1024,1024] fp32
    const int*   kps = (const int*)d_in[1];     // [16,512,2] int32
    float*       out = (float*)d_out;           // [512,3,32,32] fp32

    // Reference uses only the LAST batch entry of both inputs.
    const float* img_last = p   + (size_t)(PB - 1) * PC * PH * PW;
    const int*   kps_last = kps + (size_t)(PB - 1) * PN * 2;

    stn_patch_gather<<<PN, 256, 0, stream>>>(img_last, kps_last, out);
}